// ChebyNet_28424093565730
// MI455X (gfx1250) — compile-verified
//
#include <hip/hip_runtime.h>

typedef __attribute__((ext_vector_type(2))) float v2f;
typedef __attribute__((ext_vector_type(8))) float v8f;

#define NNODES 100000
#define NEDGES 1600000
#define DIM    128
#define UNITS  64
#define CLS    40
#define KSTEPS (DIM / 4)                  // 32 WMMA k-steps per order

// LDS fragment store: [ord(3)][ks(32)][t(3)][lane(32)][pair(2)] floats
#define FR_FLOATS (3 * KSTEPS * 3 * 32 * 2)   // 18432
#define LDS_FLOATS (FR_FLOATS + 48)           // + zero-padded bias

// workspace layout in floats (total ~27.32M floats = ~110 MB)
#define WS_DEG 0
#define WS_WN  (WS_DEG + NNODES)          // 100000
#define WS_T1  1700000                    // = WS_WN + NEDGES, 16-float aligned
#define WS_T2  14500000                   // = WS_T1 + NNODES*DIM
#define WS_WP  27300000                   // = WS_T2 + NNODES*DIM ; 3*128*40 floats
#define WS_BP  (WS_WP + 3*DIM*CLS)        // 40 floats, contiguous after Wp

// ---------------- init: zero deg, zero T1, T2 = -x -----------------------
__global__ void cheb_init_kernel(const float* __restrict__ x, float* __restrict__ ws) {
    int i = blockIdx.x * blockDim.x + threadIdx.x;
    if (i < NNODES * DIM) {
        ws[WS_T1 + i] = 0.0f;
        ws[WS_T2 + i] = -x[i];
    }
    if (i < NNODES) ws[WS_DEG + i] = 0.0f;
}

// ---------------- degree at dst ------------------------------------------
__global__ void cheb_deg_kernel(const int* __restrict__ dst, const float* __restrict__ ew,
                                float* __restrict__ deg) {
    int e = blockIdx.x * blockDim.x + threadIdx.x;
    if (e < NEDGES) unsafeAtomicAdd(&deg[dst[e]], ew[e]);
}

// ---------------- w_norm = ew / max(deg[dst], 1e-12) ---------------------
__global__ void cheb_wnorm_kernel(const int* __restrict__ dst, const float* __restrict__ ew,
                                  const float* __restrict__ deg, float* __restrict__ wn) {
    int e = blockIdx.x * blockDim.x + threadIdx.x;
    if (e < NEDGES) wn[e] = ew[e] / fmaxf(deg[dst[e]], 1e-12f);
}

// ---------------- edge-parallel SpMM: Tout[dst] += scale*wn * Tin[src] ---
// one wave per edge, one float4 (16B) per lane covers the 128-float row
__global__ void cheb_spmm_kernel(const float* __restrict__ Tin, float* __restrict__ Tout,
                                 const int* __restrict__ src, const int* __restrict__ dst,
                                 const float* __restrict__ wn, float scale) {
    long long gid = (long long)blockIdx.x * blockDim.x + threadIdx.x;
    int e    = (int)(gid >> 5);
    int lane = (int)(gid & 31);
    if (e >= NEDGES) return;
    int s = src[e], d = dst[e];
    float w = scale * wn[e];
    float4 v = ((const float4*)(Tin + (size_t)s * DIM))[lane];
    float* drow = Tout + (size_t)d * DIM + lane * 4;
    unsafeAtomicAdd(drow + 0, w * v.x);
    unsafeAtomicAdd(drow + 1, w * v.y);
    unsafeAtomicAdd(drow + 2, w * v.z);
    unsafeAtomicAdd(drow + 3, w * v.w);
}

// ---------------- fold weights: Wp[k] = W[k] @ Wd ; bp = b @ Wd + bd -----
__global__ void cheb_wfold_kernel(const float* __restrict__ W, const float* __restrict__ b,
                                  const float* __restrict__ Wd, const float* __restrict__ bd,
                                  float* __restrict__ wp) {
    int tid = blockIdx.x * blockDim.x + threadIdx.x;
    if (tid < 3 * DIM * CLS) {
        int ord = tid / (DIM * CLS);
        int r   = tid - ord * (DIM * CLS);
        int d   = r / CLS;
        int c   = r - d * CLS;
        const float* wrow = W + (size_t)ord * DIM * UNITS + (size_t)d * UNITS;
        float sum = 0.0f;
        for (int u = 0; u < UNITS; ++u) sum += wrow[u] * Wd[u * CLS + c];
        wp[tid] = sum;
    } else if (tid < 3 * DIM * CLS + CLS) {
        int c = tid - 3 * DIM * CLS;
        float sum = bd[c];
        for (int u = 0; u < UNITS; ++u) sum += b[u] * Wd[u * CLS + c];
        wp[tid] = sum;
    }
}

// ---------------- fused output GEMM via V_WMMA_F32_16X16X4_F32 -----------
// out[N,40] = x@Wp0 + T1@Wp1 + T2@Wp2 + bp
// B fragments pre-swizzled in LDS to exact WMMA lane order: each lane's
// (K,K+1) pair is one aligned ds_load_b64, no guards, no register shuffles.
__global__ void __launch_bounds__(256)
cheb_out_kernel(const float* __restrict__ x, const float* __restrict__ ws,
                float* __restrict__ out) {
    __shared__ float lds[LDS_FLOATS];             // 73.9 KB (320 KB/WGP available)
    const float* wpg = ws + WS_WP;

    // Stage folded weights, swizzled into fragment order; zero-pad cols 40..47.
    for (int i = threadIdx.x; i < FR_FLOATS; i += 256) {
        int p    = i & 1;                          // which of the lane's K pair
        int l    = (i >> 1) & 31;                  // lane
        int f    = i >> 6;                         // fragment id
        int t    = f % 3;                          // column tile
        int g    = f / 3;
        int ks   = g & (KSTEPS - 1);               // k-step
        int ord  = g >> 5;                         // Chebyshev order
        int kk   = ks * 4 + 2 * (l >> 4) + p;      // B layout: half-wave = K pair
        int n    = t * 16 + (l & 15);              // column
        lds[i] = (n < CLS) ? wpg[(ord * DIM + kk) * CLS + n] : 0.0f;
    }
    for (int i = threadIdx.x; i < 48; i += 256)    // bias, zero-padded
        lds[FR_FLOATS + i] = (i < CLS) ? wpg[3 * DIM * CLS + i] : 0.0f;
    __syncthreads();

    int wave = threadIdx.x >> 5;
    int lane = threadIdx.x & 31;
    int tile = blockIdx.x * 8 + wave;
    if (tile >= NNODES / 16) return;               // wave-uniform: EXEC stays all-1s

    int row0 = tile << 4;
    int half = lane >> 4;                          // A: lanes 16-31 hold K2/K3
    int m    = lane & 15;                          // A: row ; B/D: column

    v8f acc[3] = {v8f{}, v8f{}, v8f{}};

    const float* T1 = ws + WS_T1;
    const float* T2 = ws + WS_T2;

    #pragma unroll
    for (int ord = 0; ord < 3; ++ord) {
        const float* T = (ord == 0) ? x : (ord == 1) ? T1 : T2;
        const float* arow = T + (size_t)(row0 + m) * DIM + 2 * half;
        const float* frag = lds + (ord * KSTEPS * 3) * 64 + lane * 2;
        for (int ks = 0; ks < KSTEPS; ++ks) {
            v2f a = *(const v2f*)(arow + ks * 4);  // A 16x4 f32 fragment (2 VGPRs)
            #pragma unroll
            for (int t = 0; t < 3; ++t) {
                v2f bf = *(const v2f*)(frag + (ks * 3 + t) * 64);  // ds_load_b64
                acc[t] = __builtin_amdgcn_wmma_f32_16x16x4_f32(
                    /*neg_a=*/false, a, /*neg_b=*/false, bf,
                    /*c_mod=*/(short)0, acc[t], /*reuse_a=*/false, /*reuse_b=*/false);
            }
        }
    }

    const float* bp = lds + FR_FLOATS;
    #pragma unroll
    for (int t = 0; t < 3; ++t) {
        int n = t * 16 + m;
        float bias = bp[n];                        // padded, always in-range
        if (n < CLS) {                             // guard only the store
            #pragma unroll
            for (int r = 0; r < 8; ++r) {
                int row = row0 + r + 8 * half;     // C/D layout: M = r + 8*half
                out[(size_t)row * CLS + n] = acc[t][r] + bias;
            }
        }
    }
}

extern "C" void kernel_launch(void* const* d_in, const int* in_sizes, int n_in,
                              void* d_out, int out_size, void* d_ws, size_t ws_size,
                              hipStream_t stream) {
    const float* x  = (const float*)d_in[0];
    const int*   ei = (const int*)d_in[1];
    const float* ew = (const float*)d_in[2];
    const float* W  = (const float*)d_in[3];
    const float* b  = (const float*)d_in[4];
    const float* Wd = (const float*)d_in[5];
    const float* bd = (const float*)d_in[6];
    float* out = (float*)d_out;
    float* ws  = (float*)d_ws;

    const int* src = ei;            // edge_index[0]
    const int* dst = ei + NEDGES;   // edge_index[1]

    // 1) zero deg/T1, T2 = -x
    cheb_init_kernel<<<(NNODES * DIM + 255) / 256, 256, 0, stream>>>(x, ws);
    // 2) degree
    cheb_deg_kernel<<<(NEDGES + 255) / 256, 256, 0, stream>>>(dst, ew, ws + WS_DEG);
    // 3) normalized edge weights
    cheb_wnorm_kernel<<<(NEDGES + 255) / 256, 256, 0, stream>>>(dst, ew, ws + WS_DEG, ws + WS_WN);
    // 4) fold W@Wd and b@Wd+bd (tiny)
    cheb_wfold_kernel<<<(3 * DIM * CLS + CLS + 255) / 256, 256, 0, stream>>>(W, b, Wd, bd, ws + WS_WP);
    // 5) T1 = -A_hat x
    long long spmm_threads = (long long)NEDGES * 32;
    int spmm_blocks = (int)((spmm_threads + 255) / 256);
    cheb_spmm_kernel<<<spmm_blocks, 256, 0, stream>>>(x, ws + WS_T1, src, dst, ws + WS_WN, -1.0f);
    // 6) T2 = -2 A_hat T1 - x   (T2 pre-initialized to -x)
    cheb_spmm_kernel<<<spmm_blocks, 256, 0, stream>>>(ws + WS_T1, ws + WS_T2, src, dst, ws + WS_WN, -2.0f);
    // 7) fused WMMA output GEMM
    cheb_out_kernel<<<(NNODES / 16 + 7) / 8, 256, 0, stream>>>(x, ws, out);
}